// KMISCoarsening_21328807592274
// MI455X (gfx1250) — compile-verified
//
#include <hip/hip_runtime.h>
#include <hip/hip_bf16.h>
#include <stdint.h>
#include <stddef.h>

typedef float        f32x4 __attribute__((ext_vector_type(4)));
typedef unsigned int u32x4 __attribute__((ext_vector_type(4)));
typedef int          i32x8 __attribute__((ext_vector_type(8)));
typedef int          i32x4 __attribute__((ext_vector_type(4)));

#ifndef __has_builtin
#define __has_builtin(x) 0
#endif
#if __has_builtin(__builtin_amdgcn_tensor_load_to_lds) && __has_builtin(__builtin_amdgcn_s_wait_tensorcnt)
#define KMIS_HAVE_TDM 1
#else
#define KMIS_HAVE_TDM 0
#endif

static constexpr int MIS_ITERS = 48;   // >> ~2*log2(10240) rounds for random ranks
static constexpr int POOL_TILE = 64;   // 64 x 128 f32 = 32KB LDS tile per workgroup

// ---------------- wave32 primitives (gfx1250 is wave32) ----------------
__device__ __forceinline__ float kmisWaveSum(float v) {
#pragma unroll
  for (int o = 16; o > 0; o >>= 1) v += __shfl_xor(v, o, 32);
  return v;
}
__device__ __forceinline__ int kmisWaveMin(int v) {
#pragma unroll
  for (int o = 16; o > 0; o >>= 1) { int w = __shfl_xor(v, o, 32); v = (w < v) ? w : v; }
  return v;
}

// ---------------- init ----------------
__global__ void kmis_init(const int* rank, int n, int* nb, int* mr, int* mis,
                          int* maskb, int* cnt, int* cursor, float* degw,
                          float* counts, int* ctrl) {
  int i = blockIdx.x * blockDim.x + threadIdx.x;
  if (i < n) {
    mr[i] = rank[i]; nb[i] = n; mis[i] = 0; maskb[i] = 0;
    cnt[i] = 0; cursor[i] = 0; degw[i] = 0.f; counts[i] = 0.f;
  }
  if (i == 0) { ctrl[0] = 0; ctrl[1] = 0; ctrl[2] = 0; }
}

// weighted degree (reference: deg = scatter-add edge_attr by row) + row counts
__global__ void kmis_deg(const int* row, const float* attr, int e, float* degw, int* cnt) {
  int t = blockIdx.x * blockDim.x + threadIdx.x;
  if (t < e) {
    int r = row[t];
    atomicAdd(&degw[r], attr[t]);
    atomicAdd(&cnt[r], 1);
  }
}

// ---------------- MIS iteration (4 kernels / round, flag early-out) ----------------
__global__ void kmis_edge_min(const int* row, const int* col, int e,
                              const int* mr, int* nb, const int* ctrl) {
  if (ctrl[0]) return;
  int t = blockIdx.x * blockDim.x + threadIdx.x;
  if (t < e) atomicMin(&nb[col[t]], mr[row[t]]);
}

__global__ void kmis_node_a(const int* rank, int n, int* nb, int* mr,
                            int* mis, int* maskb, int* ctrl) {
  if (ctrl[0]) return;
  int i = blockIdx.x * blockDim.x + threadIdx.x;
  if (i < n) {
    int v = min(nb[i], mr[i]);
    mr[i] = v;
    int m = mis[i];
    if (rank[i] == v) m = 1;
    mis[i] = m;
    maskb[i] = m;   // init for coverage max-propagation
    nb[i] = n;      // reset for next round
  }
  if (i == 0) ctrl[1] = 0;  // pending counter
}

__global__ void kmis_edge_max(const int* row, const int* col, int e,
                              const int* mis, int* maskb, const int* ctrl) {
  if (ctrl[0]) return;
  int t = blockIdx.x * blockDim.x + threadIdx.x;
  if (t < e) { if (mis[row[t]]) atomicOr(&maskb[col[t]], 1); }
}

__global__ void kmis_node_b(const int* rank, int n, const int* maskb, int* mr, int* ctrl) {
  if (ctrl[0]) return;
  int i = blockIdx.x * blockDim.x + threadIdx.x;
  if (i < n) {
    int m = maskb[i];
    mr[i] = m ? n : rank[i];
    if (!m) atomicAdd(&ctrl[1], 1);
  }
  __syncthreads();
  if (threadIdx.x == 0) {
    __threadfence();
    int t = atomicAdd(&ctrl[2], 1);
    if (t == (int)gridDim.x - 1) {          // last block: convergence check
      atomicExch(&ctrl[2], 0);
      int pend = atomicAdd(&ctrl[1], 0);
      if (pend == 0) ctrl[0] = 1;           // converged -> later kernels no-op
    }
  }
}

// ---------------- CSR build: exclusive scan (single WG, wave32 scans) ----------------
__global__ void __launch_bounds__(1024) kmis_scan(const int* cnt, int* startA, int n) {
  __shared__ int wsums[32];
  __shared__ int carry;
  if (threadIdx.x == 0) carry = 0;
  __syncthreads();
  for (int base = 0; base < n; base += 1024) {
    int idx = base + (int)threadIdx.x;
    int v = (idx < n) ? cnt[idx] : 0;
    int lane = threadIdx.x & 31, wid = threadIdx.x >> 5;
    int xv = v;
#pragma unroll
    for (int o = 1; o < 32; o <<= 1) { int y = __shfl_up(xv, o, 32); if (lane >= o) xv += y; }
    if (lane == 31) wsums[wid] = xv;
    __syncthreads();
    if (wid == 0) {
      int s = wsums[lane];
#pragma unroll
      for (int o = 1; o < 32; o <<= 1) { int y = __shfl_up(s, o, 32); if (lane >= o) s += y; }
      wsums[lane] = s;
    }
    __syncthreads();
    int pre = (wid > 0) ? wsums[wid - 1] : 0;
    int incl = xv + pre + carry;
    if (idx < n) startA[idx] = incl - v;   // exclusive
    __syncthreads();
    if (threadIdx.x == 1023) carry = incl;
    __syncthreads();
  }
  if (threadIdx.x == 0) startA[n] = carry;
}

__global__ void kmis_fill_csr(const int* row, int e, const int* startA, int* cursor, int* csr) {
  int t = blockIdx.x * blockDim.x + threadIdx.x;
  if (t < e) {
    int r = row[t];
    int p = atomicAdd(&cursor[r], 1);
    csr[startA[r] + p] = t;
  }
}

// ---------------- zero-fill 425MB of output: nontemporal B128 stream ----------------
__global__ void kmis_zero(f32x4* p, long long n4) {
  long long i = (long long)blockIdx.x * blockDim.x + threadIdx.x;
  long long st = (long long)gridDim.x * blockDim.x;
  f32x4 z = {0.f, 0.f, 0.f, 0.f};
  for (; i < n4; i += st) __builtin_nontemporal_store(z, p + i);
}

// ---------------- cluster sampling: one wave per row ----------------
// value(j): j<m -> edge entry (col, 0.5*attr/degw masked by mis); j==m -> diagonal (i, 0.5*mis[i])
__global__ void __launch_bounds__(256) kmis_cluster(
    const float* attr, const float* u, const int* colA, const int* startA,
    const int* cnt, const int* mis, const float* degw, const int* csr,
    int n, int* clusterI, float* counts) {
  int lane = threadIdx.x & 31;
  int i = (blockIdx.x * blockDim.x + threadIdx.x) >> 5;
  if (i >= n) return;
  int s0 = startA[i];
  int m = cnt[i];
  int mtot = m + 1;
  float dw = degw[i]; if (dw == 0.f) dw = 1.f;
  float inv = 0.5f / dw;                      // (1-EPS)/deg
  float tot = 0.f;
  for (int j = lane; j < mtot; j += 32) {
    float v;
    if (j < m) { int e2 = csr[s0 + j]; int c = colA[e2]; v = mis[c] ? attr[e2] * inv : 0.f; }
    else       { v = mis[i] ? 0.5f : 0.f; }
    tot += v;
  }
  tot = kmisWaveSum(tot);
  int clus = 0;
  if (tot > 0.f) {
    float thr = u[i] * tot;                   // normalization cancels in cdf compare
    float acc = 0.f; int cur = -1, last = -1; clus = -1;
    for (;;) {
      int cm = 0x7fffffff;
      for (int j = lane; j < mtot; j += 32) {
        int c; float v;
        if (j < m) { int e2 = csr[s0 + j]; c = colA[e2]; v = mis[c] ? attr[e2] * inv : 0.f; }
        else       { c = i; v = mis[i] ? 0.5f : 0.f; }
        if (v > 0.f && c > cur && c < cm) cm = c;
      }
      cm = kmisWaveMin(cm);
      if (cm == 0x7fffffff) { clus = (last < 0) ? 0 : last; break; }
      float sv = 0.f;
      for (int j = lane; j < mtot; j += 32) {
        int c; float v;
        if (j < m) { int e2 = csr[s0 + j]; c = colA[e2]; v = mis[c] ? attr[e2] * inv : 0.f; }
        else       { c = i; v = mis[i] ? 0.5f : 0.f; }
        if (c == cm) sv += v;
      }
      sv = kmisWaveSum(sv);
      acc += sv;                              // cumsum in column order (matches dense cdf)
      if (acc > thr) { clus = cm; break; }
      cur = cm; last = cm;
    }
  }
  if (lane == 0) {
    clusterI[i] = clus;
    atomicAdd(&counts[clus], 1.0f);
  }
}

// ---------------- pooling: TDM stages a 64x128 x-tile into LDS, then scatter ----------------
__global__ void __launch_bounds__(256) kmis_pool(const float* x, const int* clusterI,
                                                 float* outx, int n, int d) {
  __shared__ float xs[POOL_TILE * 128];
  int node0 = blockIdx.x * POOL_TILE;
  int rows = n - node0; if (rows > POOL_TILE) rows = POOL_TILE;
  if (rows <= 0) return;
#if KMIS_HAVE_TDM
  if (d == 128) {
    if (threadIdx.x < 32) {   // wave 0 issues the Tensor DMA (per-wave op)
      unsigned lds_b = (unsigned)(size_t)(&xs[0]);  // flat addr low 32b = LDS byte offset
      unsigned long long ga = (unsigned long long)(size_t)(x + (size_t)node0 * 128);
      u32x4 g0;
      g0[0] = 1u;                                          // count=1 valid descriptor
      g0[1] = lds_b;                                       // lds_addr
      g0[2] = (unsigned)(ga & 0xFFFFFFFFull);              // global_addr[31:0]
      g0[3] = (unsigned)((ga >> 32) & 0x01FFFFFFull) | (2u << 30); // addr[56:32] | type=2
      unsigned td0 = 128u, td1 = (unsigned)rows;           // tensor dims (elements)
      unsigned t0 = 128u, t1 = (unsigned)rows, st0 = 128u; // tile dims, dim0 stride
      i32x8 g1;
      g1[0] = (int)(2u << 16);                             // data_size=2 -> 4 bytes
      g1[1] = (int)((td0 & 0xFFFFu) << 16);                // tensor_dim0 lo16
      g1[2] = (int)((td0 >> 16) | ((td1 & 0xFFFFu) << 16));// tensor_dim0 hi | dim1 lo
      g1[3] = (int)((td1 >> 16) | (t0 << 16));             // tensor_dim1 hi | tile_dim0
      g1[4] = (int)t1;                                     // tile_dim1 (tile_dim2=0)
      g1[5] = (int)st0;                                    // tensor_dim0_stride lo32
      g1[6] = 0;
      g1[7] = 0;
      i32x4 g2; g2[0] = g2[1] = g2[2] = g2[3] = 0;         // 2D tensor: groups 2/3 zero
      i32x4 g3 = g2;
      i32x8 g4;                                            // 6-arg toolchain form: extra group
      g4[0] = g4[1] = g4[2] = g4[3] = g4[4] = g4[5] = g4[6] = g4[7] = 0;
      __builtin_amdgcn_tensor_load_to_lds(g0, g1, g2, g3, g4, 0);
      __builtin_amdgcn_s_wait_tensorcnt(0);
    }
    __syncthreads();
    for (int idx = threadIdx.x; idx < rows * 128; idx += blockDim.x) {
      int r = idx >> 7, c = idx & 127;
      int cl = clusterI[node0 + r];
      atomicAdd(&outx[(size_t)cl * 128 + c], xs[idx]);
    }
    return;
  }
#endif
  for (int idx = threadIdx.x; idx < rows * d; idx += blockDim.x) {
    int r = idx / d, c = idx - r * d;
    int cl = clusterI[node0 + r];
    atomicAdd(&outx[(size_t)cl * d + c], x[(size_t)(node0 + r) * d + c]);
  }
}

__global__ void kmis_norm(float* outx, const float* counts, int n, int d) {
  long long t = (long long)blockIdx.x * blockDim.x + threadIdx.x;
  long long tot = (long long)n * d;
  if (t < tot) {
    float c = counts[t / d];
    if (c < 1.f) c = 1.f;
    outx[t] /= c;
  }
}

__global__ void kmis_adj(const int* row, const int* col, const float* attr, int e,
                         const int* clusterI, float* adj, int n) {
  int t = blockIdx.x * blockDim.x + threadIdx.x;
  if (t < e) {
    size_t idx = (size_t)clusterI[row[t]] * (size_t)n + (size_t)clusterI[col[t]];
    atomicAdd(&adj[idx], attr[t]);
  }
}

__global__ void kmis_tail(const int* mis, const int* clusterI, int n, float* misf, float* clf) {
  int i = blockIdx.x * blockDim.x + threadIdx.x;
  if (i < n) {
    misf[i] = mis[i] ? 1.f : 0.f;
    clf[i] = (float)clusterI[i];
  }
}

// ---------------- host launcher ----------------
extern "C" void kernel_launch(void* const* d_in, const int* in_sizes, int n_in,
                              void* d_out, int out_size, void* d_ws, size_t ws_size,
                              hipStream_t stream) {
  const float* x    = (const float*)d_in[0];
  const float* attr = (const float*)d_in[1];
  const float* u    = (const float*)d_in[2];
  const int*   ei   = (const int*)d_in[3];
  const int*   rank = (const int*)d_in[4];
  const int n = in_sizes[4];
  const int d = in_sizes[0] / n;
  const int e = in_sizes[1];
  const int* row = ei;
  const int* col = ei + e;

  float* outx = (float*)d_out;
  float* adj  = outx + (size_t)n * d;
  float* misf = adj + (size_t)n * n;
  float* clf  = misf + n;

  // workspace carve-out (256B aligned slices)
  char* w = (char*)d_ws;
  auto carve = [&](size_t bytes) -> char* {
    char* p = w; w += (bytes + 255) & ~(size_t)255; return p;
  };
  int*   nbv    = (int*)carve((size_t)n * 4);
  int*   mr     = (int*)carve((size_t)n * 4);
  int*   mis    = (int*)carve((size_t)n * 4);
  int*   maskb  = (int*)carve((size_t)n * 4);
  int*   cnt    = (int*)carve((size_t)n * 4);
  int*   startA = (int*)carve((size_t)(n + 1) * 4);
  int*   cursor = (int*)carve((size_t)n * 4);
  int*   clustI = (int*)carve((size_t)n * 4);
  int*   csr    = (int*)carve((size_t)e * 4);
  float* degw   = (float*)carve((size_t)n * 4);
  float* counts = (float*)carve((size_t)n * 4);
  int*   ctrl   = (int*)carve(4 * 4);

  const int gN = (n + 255) / 256;
  const int gE = (e + 255) / 256;

  kmis_init<<<gN, 256, 0, stream>>>(rank, n, nbv, mr, mis, maskb, cnt, cursor, degw, counts, ctrl);
  kmis_deg<<<gE, 256, 0, stream>>>(row, attr, e, degw, cnt);

  for (int it = 0; it < MIS_ITERS; ++it) {
    kmis_edge_min<<<gE, 256, 0, stream>>>(row, col, e, mr, nbv, ctrl);
    kmis_node_a<<<gN, 256, 0, stream>>>(rank, n, nbv, mr, mis, maskb, ctrl);
    kmis_edge_max<<<gE, 256, 0, stream>>>(row, col, e, mis, maskb, ctrl);
    kmis_node_b<<<gN, 256, 0, stream>>>(rank, n, maskb, mr, ctrl);
  }

  kmis_scan<<<1, 1024, 0, stream>>>(cnt, startA, n);
  kmis_fill_csr<<<gE, 256, 0, stream>>>(row, e, startA, cursor, csr);

  // zero out + adj regions (425MB stream, NT B128 stores)
  long long n4 = ((long long)n * d + (long long)n * n) / 4;
  kmis_zero<<<32768, 256, 0, stream>>>((f32x4*)outx, n4);

  kmis_cluster<<<(n + 7) / 8, 256, 0, stream>>>(attr, u, col, startA, cnt, mis, degw, csr,
                                                n, clustI, counts);
  kmis_pool<<<(n + POOL_TILE - 1) / POOL_TILE, 256, 0, stream>>>(x, clustI, outx, n, d);
  {
    long long tot = (long long)n * d;
    int g = (int)((tot + 255) / 256);
    kmis_norm<<<g, 256, 0, stream>>>(outx, counts, n, d);
  }
  kmis_adj<<<gE, 256, 0, stream>>>(row, col, attr, e, clustI, adj, n);
  kmis_tail<<<gN, 256, 0, stream>>>(mis, clustI, n, misf, clf);
}